// Propagation_prob2_64974265254028
// MI455X (gfx1250) — compile-verified
//
#include <hip/hip_runtime.h>
#include <hip/hip_bf16.h>
#include <stdint.h>

#ifndef __has_builtin
#define __has_builtin(x) 0
#endif

#if __has_builtin(__builtin_amdgcn_global_load_async_to_lds_b32)
#define HAVE_ASYNC_LDS 1
#else
#define HAVE_ASYNC_LDS 0
#endif

typedef float v4f __attribute__((ext_vector_type(4)));
typedef __attribute__((address_space(1))) int* gptr_i32;
typedef __attribute__((address_space(3))) int* lptr_i32;

#define TW 256                 // tile width (pixels)
#define TH 4                   // tile height (rows)
#define HALO 2
#define LDS_W (TW + 2 * HALO)  // 260
#define LDS_H (TH + 2 * HALO)  // 8
#define LDS_STRIDE (LDS_W + 4) // 264, pad for banking

// Output channels, offsets relative to center (2,2):
// c0 = S(-2,-2)+S(-2,0)   c1 = S(-1,-1)+S(0,-2)   c2 = S(0,0)
// c3 = S(0,-2)+S(0,2)     c4 = S(-2,0)+S(2,0)     c5 = S(-1,-1)
// c6 = S(1,1)             c7 = S(-1,1)            c8 = S(1,-1)
__global__ __launch_bounds__(256, 2) void prop9_stencil_kernel(
    const float* __restrict__ in, float* __restrict__ out,
    int D, int H, int W) {
  __shared__ float tile[LDS_H * LDS_STRIDE];

  const int tid = threadIdx.x;
  const int w0 = blockIdx.x * TW;
  const int h0 = blockIdx.y * TH;
  const int nd = blockIdx.z;                 // combined n*D + d
  const size_t plane = (size_t)H * W;
  const float* __restrict__ src = in + (size_t)nd * plane;

  // ---- Stage halo tile (8 x 260) into LDS with edge clamp ----
  // gfx1250 path: GLOBAL_LOAD_ASYNC_TO_LDS_B32 (ASYNCcnt-tracked DMA into LDS)
  for (int idx = tid; idx < LDS_H * LDS_W; idx += 256) {
    const int r = idx / LDS_W;
    const int c = idx - r * LDS_W;
    int gh = h0 + r - HALO; gh = gh < 0 ? 0 : (gh > H - 1 ? H - 1 : gh);
    int gw = w0 + c - HALO; gw = gw < 0 ? 0 : (gw > W - 1 ? W - 1 : gw);
    const float* gp = src + (size_t)gh * W + gw;
    float* lp = &tile[r * LDS_STRIDE + c];
#if HAVE_ASYNC_LDS
    __builtin_amdgcn_global_load_async_to_lds_b32(
        (gptr_i32)(gp),
        (lptr_i32)(lp),
        /*offset=*/0, /*cpol=*/0);
#else
    *lp = *gp;
#endif
  }
#if HAVE_ASYNC_LDS
#if __has_builtin(__builtin_amdgcn_s_wait_asynccnt)
  __builtin_amdgcn_s_wait_asynccnt(0);
#else
  asm volatile("s_wait_asynccnt 0" ::: "memory");
#endif
#endif
  __syncthreads();

  // ---- Compute: each thread produces a float4 per channel ----
  const int ty = tid >> 6;        // 0..3   (row within tile)
  const int tx = tid & 63;        // 0..63  (quad within row)
  const int lw = tx << 2;         // 0..252 (start col of this thread's 4 pixels)
  const float* t = &tile[(ty + HALO) * LDS_STRIDE + (lw + HALO)];

  v4f c0, c1, c2, c3, c4, c5, c6, c7, c8;
#pragma unroll
  for (int j = 0; j < 4; ++j) {
    const float am2m2 = t[-2 * LDS_STRIDE - 2 + j];
    const float am20  = t[-2 * LDS_STRIDE + 0 + j];
    const float am1m1 = t[-1 * LDS_STRIDE - 1 + j];
    const float am11  = t[-1 * LDS_STRIDE + 1 + j];
    const float a0m2  = t[-2 + j];
    const float a00   = t[0 + j];
    const float a02   = t[2 + j];
    const float a1m1  = t[1 * LDS_STRIDE - 1 + j];
    const float a11   = t[1 * LDS_STRIDE + 1 + j];
    const float a20   = t[2 * LDS_STRIDE + 0 + j];
    c0[j] = am2m2 + am20;
    c1[j] = am1m1 + a0m2;
    c2[j] = a00;
    c3[j] = a0m2 + a02;
    c4[j] = am20 + a20;
    c5[j] = am1m1;
    c6[j] = a11;
    c7[j] = am11;
    c8[j] = a1m1;
  }

  // ---- 9 x 128-bit non-temporal stores (write-once stream, bypass L2 retention)
  const int n = nd / D;
  const int d = nd - n * D;
  const int h = h0 + ty;
  const size_t ch_stride = (size_t)D * plane;           // 48*256*512 floats
  float* obase = out + ((size_t)(n * 9) * D + d) * plane +
                 (size_t)h * W + (size_t)(w0 + lw);

  __builtin_nontemporal_store(c0, (v4f*)(obase + 0 * ch_stride));
  __builtin_nontemporal_store(c1, (v4f*)(obase + 1 * ch_stride));
  __builtin_nontemporal_store(c2, (v4f*)(obase + 2 * ch_stride));
  __builtin_nontemporal_store(c3, (v4f*)(obase + 3 * ch_stride));
  __builtin_nontemporal_store(c4, (v4f*)(obase + 4 * ch_stride));
  __builtin_nontemporal_store(c5, (v4f*)(obase + 5 * ch_stride));
  __builtin_nontemporal_store(c6, (v4f*)(obase + 6 * ch_stride));
  __builtin_nontemporal_store(c7, (v4f*)(obase + 7 * ch_stride));
  __builtin_nontemporal_store(c8, (v4f*)(obase + 8 * ch_stride));
}

extern "C" void kernel_launch(void* const* d_in, const int* in_sizes, int n_in,
                              void* d_out, int out_size, void* d_ws, size_t ws_size,
                              hipStream_t stream) {
  (void)in_sizes; (void)n_in; (void)out_size; (void)d_ws; (void)ws_size;
  const float* in = (const float*)d_in[0];
  float* out = (float*)d_out;
  const int N = 2, D = 48, H = 256, W = 512;
  dim3 grid(W / TW, H / TH, N * D);   // (2, 64, 96)
  dim3 block(256, 1, 1);
  hipLaunchKernelGGL(prop9_stencil_kernel, grid, block, 0, stream, in, out, D, H, W);
}